// RobustClassifierLayer_1778116460950
// MI455X (gfx1250) — compile-verified
//
#include <hip/hip_runtime.h>

// ---------------------------------------------------------------------------
// RobustClassifierLayer for MI455X (gfx1250, wave32).
//   Phase 1: batched Gram matrix via V_WMMA_F32_16X16X4_F32 (exact f32),
//            then squared-distance + diag-subtract + clamp -> C [B,n,n].
//   Phase 2: 30 mirror-ascent steps on Z [B,K,n,n] (L2-resident, ~21 MB):
//            K1 softmax/gamma/p/cost  (per (b,k) block, wave32 shuffles)
//            K2 argmin_k p_k[j]       (global cross-k coupling point)
//            K3 analytic softmax-grad update of Z
//   Final:   K1 once more, writing p = colsum(gamma) straight into d_out.
// Everything stays f32: the argmin in the gradient is decision-sensitive,
// so bf16/fp8 WMMA would change control flow of the ascent; the f32 WMMA
// path is the precision-matching CDNA5 matrix instruction.
// ---------------------------------------------------------------------------

typedef __attribute__((ext_vector_type(2))) float v2f;
typedef __attribute__((ext_vector_type(8))) float v8f;

#define B_  32
#define K_  10
#define N_  128
#define F_  512
#define NITERS_ 30
#define LR_  0.5f
#define RHO_ 10.0f

// ---------------- Phase 1: distances --------------------------------------

__global__ void xn_kernel(const float* __restrict__ X, float* __restrict__ xn) {
  int r = blockIdx.x * blockDim.x + threadIdx.x;     // row id in [0, B*n)
  if (r >= B_ * N_) return;
  const float4* xr = (const float4*)(X + (size_t)r * F_);
  float s = 0.f;
#pragma unroll 4
  for (int i = 0; i < F_ / 4; ++i) {
    float4 v = xr[i];
    s += v.x * v.x + v.y * v.y + v.z * v.z + v.w * v.w;
  }
  xn[r] = s;
}

// One wave (32 threads) per 16x16 tile of G = X X^T for one batch.
// raw[m][n] = xn[m] + xn[n] - 2*G[m][n]
__global__ void gram_dist_kernel(const float* __restrict__ X,
                                 const float* __restrict__ xn,
                                 float* __restrict__ rawD) {
  const int tn   = blockIdx.x;          // column tile 0..7
  const int tm   = blockIdx.y;          // row tile    0..7
  const int b    = blockIdx.z;          // batch       0..31
  const int lane = threadIdx.x;         // 0..31 (wave32)
  const int l16  = lane & 15;
  const int half = lane >> 4;

  const float* Xb   = X + (size_t)b * N_ * F_;
  const float* arow = Xb + (size_t)(tm * 16 + l16) * F_;   // A: rows of X
  const float* brow = Xb + (size_t)(tn * 16 + l16) * F_;   // B = X^T: cols

  v8f acc = {};
  for (int k = 0; k < F_; k += 4) {
    // 32-bit A 16x4 layout: lanes 0-15 hold K={0,1}, lanes 16-31 K={2,3}
    v2f a, bb;
    a.x  = arow[k + 2 * half];
    a.y  = arow[k + 2 * half + 1];
    bb.x = brow[k + 2 * half];
    bb.y = brow[k + 2 * half + 1];
    acc = __builtin_amdgcn_wmma_f32_16x16x4_f32(
        /*neg_a=*/false, a, /*neg_b=*/false, bb,
        /*c_mod=*/(short)0, acc, /*reuse_a=*/false, /*reuse_b=*/false);
  }

  // C/D layout: acc[r] -> (M = r + 8*half, N = l16)
  const float xncol = xn[b * N_ + tn * 16 + l16];
#pragma unroll
  for (int r = 0; r < 8; ++r) {
    int m = tm * 16 + r + 8 * half;
    float val = xn[b * N_ + m] + xncol - 2.0f * acc[r];
    rawD[(size_t)b * N_ * N_ + (size_t)m * N_ + tn * 16 + l16] = val;
  }
}

// C[b][i][j] = max(raw[b][i][j] - raw[b][j][j], 0)   (diag != 0 numerically)
__global__ void cost_matrix_kernel(const float* __restrict__ rawD,
                                   float* __restrict__ C) {
  int idx = blockIdx.x * blockDim.x + threadIdx.x;
  if (idx >= B_ * N_ * N_) return;
  int j = idx % N_;
  int b = idx / (N_ * N_);
  float d = rawD[idx] - rawD[(size_t)b * N_ * N_ + (size_t)j * N_ + j];
  C[idx] = d > 0.f ? d : 0.f;
}

__global__ void zero_kernel(float* __restrict__ p, int n) {
  int i = blockIdx.x * blockDim.x + threadIdx.x;
  if (i < n) p[i] = 0.f;
}

// ---------------- Phase 2: mirror ascent ----------------------------------

__device__ __forceinline__ float wave_max(float v) {
#pragma unroll
  for (int off = 16; off >= 1; off >>= 1) v = fmaxf(v, __shfl_xor(v, off, 32));
  return v;
}
__device__ __forceinline__ float wave_sum(float v) {
#pragma unroll
  for (int off = 16; off >= 1; off >>= 1) v += __shfl_xor(v, off, 32);
  return v;
}

// Block per (b,k): 256 threads = 8 waves, wave w owns rows [16w, 16w+16).
// Produces gamma = Q * softmax(Z), p = colsum(gamma), cost = <gamma, C>.
__global__ void softmax_gamma_kernel(const float* __restrict__ Z,
                                     const float* __restrict__ Q,
                                     const float* __restrict__ Cm,
                                     float* __restrict__ gamma,
                                     float* __restrict__ p_out,
                                     float* __restrict__ cost_out) {
  const int bk = blockIdx.x;            // 0..319
  const int b  = bk / K_;
  __shared__ float p_s[N_];
  __shared__ float cost_s;
  const int t = threadIdx.x;
  if (t < N_) p_s[t] = 0.f;
  if (t == 0) cost_s = 0.f;
  __syncthreads();

  const int w = t >> 5, lane = t & 31;
  const float* Zs = Z     + (size_t)bk * N_ * N_;
  float*       Gs = gamma + (size_t)bk * N_ * N_;
  const float* Cb = Cm    + (size_t)b  * N_ * N_;
  const float* Qs = Q     + (size_t)bk * N_;

  float costAcc = 0.f;
  for (int i = 0; i < 16; ++i) {
    const int l = w * 16 + i;
    const float* zr = Zs + (size_t)l * N_;
    float z0 = zr[lane], z1 = zr[lane + 32], z2 = zr[lane + 64], z3 = zr[lane + 96];
    float m = wave_max(fmaxf(fmaxf(z0, z1), fmaxf(z2, z3)));
    float e0 = __expf(z0 - m), e1 = __expf(z1 - m);
    float e2 = __expf(z2 - m), e3 = __expf(z3 - m);
    float s = wave_sum(e0 + e1 + e2 + e3);
    float scale = Qs[l] / s;                       // row-sum == Q exactly
    float g0 = e0 * scale, g1 = e1 * scale, g2 = e2 * scale, g3 = e3 * scale;
    float* gr = Gs + (size_t)l * N_;
    gr[lane] = g0; gr[lane + 32] = g1; gr[lane + 64] = g2; gr[lane + 96] = g3;
    atomicAdd(&p_s[lane],      g0);                // ds_add_f32 column sums
    atomicAdd(&p_s[lane + 32], g1);
    atomicAdd(&p_s[lane + 64], g2);
    atomicAdd(&p_s[lane + 96], g3);
    const float* cr = Cb + (size_t)l * N_;
    costAcc += g0 * cr[lane] + g1 * cr[lane + 32] +
               g2 * cr[lane + 64] + g3 * cr[lane + 96];
  }
  atomicAdd(&cost_s, costAcc);
  __syncthreads();
  if (t < N_) p_out[(size_t)bk * N_ + t] = p_s[t];
  if (t == 0) cost_out[bk] = cost_s;
}

// argmin over k of p[b,k,j] -> the cross-k coupling of the objective's min.
__global__ void argmin_kernel(const float* __restrict__ p, int* __restrict__ amin) {
  int idx = blockIdx.x * blockDim.x + threadIdx.x;  // b*N_ + j
  if (idx >= B_ * N_) return;
  int b = idx / N_, j = idx % N_;
  float best = p[(size_t)(b * K_) * N_ + j];
  int bi = 0;
#pragma unroll
  for (int k = 1; k < K_; ++k) {
    float v = p[(size_t)(b * K_ + k) * N_ + j];
    if (v < best) { best = v; bi = k; }
  }
  amin[idx] = bi;
}

// Z += LR * gamma * (g - (1/Q_l) * sum_j gamma*g),  g = 1{argmin==k} - 2*rho*pen*C
__global__ void update_kernel(float* __restrict__ Z,
                              const float* __restrict__ gamma,
                              const float* __restrict__ Cm,
                              const float* __restrict__ Q,
                              const float* __restrict__ cost,
                              const float* __restrict__ theta,
                              const int* __restrict__ amin) {
  const int bk = blockIdx.x;
  const int b  = bk / K_, k = bk % K_;
  float pen = cost[bk] - theta[bk];
  pen = pen > 0.f ? pen : 0.f;
  const float coef = 2.f * RHO_ * pen;

  const int t = threadIdx.x, w = t >> 5, lane = t & 31;
  const int* am = amin + b * N_;
  const float i0 = (am[lane]      == k) ? 1.f : 0.f;
  const float i1 = (am[lane + 32] == k) ? 1.f : 0.f;
  const float i2 = (am[lane + 64] == k) ? 1.f : 0.f;
  const float i3 = (am[lane + 96] == k) ? 1.f : 0.f;

  float*       Zs = Z     + (size_t)bk * N_ * N_;
  const float* Gs = gamma + (size_t)bk * N_ * N_;
  const float* Cb = Cm    + (size_t)b  * N_ * N_;
  const float* Qs = Q     + (size_t)bk * N_;

  for (int i = 0; i < 16; ++i) {
    const int l = w * 16 + i;
    const float* gr = Gs + (size_t)l * N_;
    const float* cr = Cb + (size_t)l * N_;
    float*       zr = Zs + (size_t)l * N_;
    float ga = gr[lane], gb = gr[lane + 32], gc = gr[lane + 64], gd = gr[lane + 96];
    float q0 = i0 - coef * cr[lane];
    float q1 = i1 - coef * cr[lane + 32];
    float q2 = i2 - coef * cr[lane + 64];
    float q3 = i3 - coef * cr[lane + 96];
    float tot = wave_sum(ga * q0 + gb * q1 + gc * q2 + gd * q3);
    float rowdot = tot / Qs[l];                   // sum_j S[l,j]*g[l,j]
    zr[lane]      += LR_ * (ga * (q0 - rowdot));
    zr[lane + 32] += LR_ * (gb * (q1 - rowdot));
    zr[lane + 64] += LR_ * (gc * (q2 - rowdot));
    zr[lane + 96] += LR_ * (gd * (q3 - rowdot));
  }
}

// ---------------------------------------------------------------------------

extern "C" void kernel_launch(void* const* d_in, const int* in_sizes, int n_in,
                              void* d_out, int out_size, void* d_ws, size_t ws_size,
                              hipStream_t stream) {
  const float* X     = (const float*)d_in[0];   // [32,128,512]
  const float* Q     = (const float*)d_in[1];   // [32,10,128]
  const float* theta = (const float*)d_in[2];   // [32,10]
  float* out = (float*)d_out;                   // [32,10,128]

  // workspace layout (bytes); total ~46.4 MB, all L2-resident on MI455X
  char* ws = (char*)d_ws;
  float* xn    = (float*)(ws);                                   //   16 KB
  float* rawD  = (float*)(ws + (16u << 10));                     //    2 MB
  float* Cm    = (float*)(ws + (16u << 10) + 2097152u);          //    2 MB
  float* Z     = (float*)(ws + 4210688u);                        // 20.97 MB
  float* gamma = (float*)(ws + 25182208u);                       // 20.97 MB
  float* p     = (float*)(ws + 46153728u);                       //  160 KB
  float* cost  = (float*)(ws + 46317568u);                       // 1.25 KB
  int*   amin  = (int*)  (ws + 46318848u);                       //   16 KB
  (void)in_sizes; (void)n_in; (void)out_size; (void)ws_size;

  // Phase 1: C = clamp(dist - diag(dist), 0)
  xn_kernel<<<(B_ * N_ + 127) / 128, 128, 0, stream>>>(X, xn);
  gram_dist_kernel<<<dim3(8, 8, B_), 32, 0, stream>>>(X, xn, rawD);
  cost_matrix_kernel<<<(B_ * N_ * N_ + 255) / 256, 256, 0, stream>>>(rawD, Cm);

  // Phase 2: 30 ascent steps (Z0 = 0)
  zero_kernel<<<(B_ * K_ * N_ * N_ + 255) / 256, 256, 0, stream>>>(Z, B_ * K_ * N_ * N_);
  for (int it = 0; it < NITERS_; ++it) {
    softmax_gamma_kernel<<<B_ * K_, 256, 0, stream>>>(Z, Q, Cm, gamma, p, cost);
    argmin_kernel<<<(B_ * N_ + 255) / 256, 256, 0, stream>>>(p, amin);
    update_kernel<<<B_ * K_, 256, 0, stream>>>(Z, gamma, Cm, Q, cost, theta, amin);
  }

  // Final: p_hat = colsum(gamma(Z_final)) written straight to d_out
  softmax_gamma_kernel<<<B_ * K_, 256, 0, stream>>>(Z, Q, Cm, gamma, out, cost);
}